// GATLayerComplex_19172734010026
// MI455X (gfx1250) — compile-verified
//
#include <hip/hip_runtime.h>
#include <math.h>

typedef __attribute__((ext_vector_type(2))) float v2f;
typedef __attribute__((ext_vector_type(8))) float v8f;

#define B_   2
#define N_   3000
#define F_   64
#define H_   4
#define OUT_ 16
#define ATT_ 16
#define NT_  ((N_ + 15) / 16)   /* 188 column/row tiles */

// ---------------------------------------------------------------------------
// Phase 1: projections.  One thread per (b, n, head); 48 fp32 accumulators.
//   feats  [B,H,N,16] = h @ kernel
//   fself  [B,H,N,16] = h @ att_kernel
//   fother [B,H,N,16] = h @ att_kernel_2   (row-major; transposed use in S)
// Weight addresses vary per lane (hd = idx % H_) -> VMEM loads, as are h loads.
// ---------------------------------------------------------------------------
__global__ __launch_bounds__(256) void gat_project(
    const float* __restrict__ h,
    const float* __restrict__ Wk,
    const float* __restrict__ Wa1,
    const float* __restrict__ Wa2,
    float* __restrict__ feats,
    float* __restrict__ fself,
    float* __restrict__ fother)
{
  int idx = blockIdx.x * blockDim.x + threadIdx.x;
  if (idx >= B_ * N_ * H_) return;
  int hd = idx % H_;
  int bn = idx / H_;                    // b*N + n
  const float4* hp = (const float4*)(h + (size_t)bn * F_);

  float aK[16], a1[16], a2[16];
  #pragma unroll
  for (int i = 0; i < 16; ++i) { aK[i] = 0.f; a1[i] = 0.f; a2[i] = 0.f; }

  for (int f4 = 0; f4 < F_ / 4; ++f4) {
    float4 hv = hp[f4];
    float hh[4] = {hv.x, hv.y, hv.z, hv.w};
    #pragma unroll
    for (int j = 0; j < 4; ++j) {
      int f = f4 * 4 + j;
      const float4* wk = (const float4*)(Wk  + ((size_t)hd * F_ + f) * OUT_);
      const float4* w1 = (const float4*)(Wa1 + ((size_t)hd * F_ + f) * ATT_);
      const float4* w2 = (const float4*)(Wa2 + ((size_t)hd * F_ + f) * ATT_);
      #pragma unroll
      for (int q = 0; q < 4; ++q) {
        float4 kv = wk[q], v1 = w1[q], v2 = w2[q];
        aK[4*q+0] += hh[j] * kv.x;  aK[4*q+1] += hh[j] * kv.y;
        aK[4*q+2] += hh[j] * kv.z;  aK[4*q+3] += hh[j] * kv.w;
        a1[4*q+0] += hh[j] * v1.x;  a1[4*q+1] += hh[j] * v1.y;
        a1[4*q+2] += hh[j] * v1.z;  a1[4*q+3] += hh[j] * v1.w;
        a2[4*q+0] += hh[j] * v2.x;  a2[4*q+1] += hh[j] * v2.y;
        a2[4*q+2] += hh[j] * v2.z;  a2[4*q+3] += hh[j] * v2.w;
      }
    }
  }

  int b = bn / N_;
  int n = bn % N_;
  size_t off = (((size_t)b * H_ + hd) * N_ + n) * 16;
  float4* pK = (float4*)(feats  + off);
  float4* p1 = (float4*)(fself  + off);
  float4* p2 = (float4*)(fother + off);
  #pragma unroll
  for (int q = 0; q < 4; ++q) {
    pK[q] = make_float4(aK[4*q+0], aK[4*q+1], aK[4*q+2], aK[4*q+3]);
    p1[q] = make_float4(a1[4*q+0], a1[4*q+1], a1[4*q+2], a1[4*q+3]);
    p2[q] = make_float4(a2[4*q+0], a2[4*q+1], a2[4*q+2], a2[4*q+3]);
  }
}

// ---------------------------------------------------------------------------
// Phase 2: fused masked attention (flash-style), one wave per 16-row tile of
// one (b, head).  V_WMMA_F32_16X16X4_F32 chains (K=16 -> 4 wmma each) for:
//   S = fself @ fotherT          (skipped for all-zero adjacency tiles)
//   O += P @ feats_tile
//   L += P @ ones                (softmax denominators via WMMA, no shuffles)
//
// f32 WMMA operand layouts (wave32):
//   A 16x4 : lane l -> M = l&15 ; half = l>>4 selects K {0,1} / {2,3} (v2f)
//   B 4x16 : lane l -> N = l&15 ; half selects K pair                (v2f)
//   C 16x16: vgpr r, lane l -> row M = r + 8*half, col = l&15        (v8f)
//
// All branches inside the tile loop are wave-uniform (__any vote), so EXEC
// stays all-ones across every WMMA as the ISA requires.
// ---------------------------------------------------------------------------
__global__ __launch_bounds__(256) void gat_attention(
    const float* __restrict__ adj,     // [B,N,N]
    const float* __restrict__ feats,   // [B,H,N,16]
    const float* __restrict__ fself,   // [B,H,N,16]
    const float* __restrict__ fother,  // [B,H,N,16]
    const float* __restrict__ bias,    // [H,16]
    float* __restrict__ out)           // [B,N,H*16]
{
  __shared__ float pbuf[8][16][20];    // per-wave P staging, stride 20: no bank conflict

  const int lane = threadIdx.x & 31;
  const int wv   = threadIdx.x >> 5;
  const int w    = blockIdx.x * 8 + wv;          // 188*8 = 1504 = B*H*NT_
  const int b    = w / (H_ * NT_);
  const int rem  = w - b * (H_ * NT_);
  const int hd   = rem / NT_;
  const int rt   = rem % NT_;

  const int half = lane >> 4;
  const int lm   = lane & 15;

  const size_t bh  = ((size_t)b * H_ + hd) * N_;
  const float* fs  = fself  + bh * 16;
  const float* fo  = fother + bh * 16;
  const float* ft  = feats  + bh * 16;
  const float* ap  = adj + (size_t)b * N_ * N_;

  // ---- hoist A operand (f_self rows of this tile) into registers ----
  const int rowA = min(rt * 16 + lm, N_ - 1);    // clamp keeps EXEC full
  v2f aS[4];
  #pragma unroll
  for (int kk = 0; kk < 4; ++kk)
    aS[kk] = *(const v2f*)(fs + (size_t)rowA * 16 + kk * 4 + 2 * half);

  const v2f onesB = {1.f, 1.f};                  // B operand for row sums

  v8f O  = {0.f, 0.f, 0.f, 0.f, 0.f, 0.f, 0.f, 0.f};
  v8f Lm = {0.f, 0.f, 0.f, 0.f, 0.f, 0.f, 0.f, 0.f};
  float m[8];
  #pragma unroll
  for (int r = 0; r < 8; ++r) m[r] = -3.0e38f;

  for (int ct = 0; ct < NT_; ++ct) {
    const int colbase = ct * 16;
    const int cn      = colbase + lm;
    const int cnc     = min(cn, N_ - 1);
    const bool cok    = (cn < N_);

    // ---- adjacency tile first: enables all-zero fast path ----
    float av[8];
    bool anynz = false;
    #pragma unroll
    for (int r = 0; r < 8; ++r) {
      const int rn = min(rt * 16 + r + 8 * half, N_ - 1);
      float a0 = ap[(size_t)rn * N_ + cnc];
      a0 = cok ? a0 : 0.0f;
      av[r] = a0;
      anynz = anynz || (a0 != 0.f);
    }
    if (!__any(anynz)) {
      // Fully masked tile: pO would be all zero (O/L untouched); masked-exp
      // terms underflow to exact 0 in fp32 (or get wiped by scale=0 later),
      // so only the running max needs the -1e9 update.
      #pragma unroll
      for (int r = 0; r < 8; ++r) m[r] = fmaxf(m[r], -1.0e9f);
      continue;
    }

    // ---- B operand: f_otherT column tile ----
    const int cnB = min(colbase + lm, N_ - 1);
    v2f bO[4];
    #pragma unroll
    for (int kk = 0; kk < 4; ++kk)
      bO[kk] = *(const v2f*)(fo + (size_t)cnB * 16 + kk * 4 + 2 * half);

    // ---- S = fself_tile @ fother_tileT  (4x K=4 chained wmma) ----
    v8f S = {0.f, 0.f, 0.f, 0.f, 0.f, 0.f, 0.f, 0.f};
    #pragma unroll
    for (int kk = 0; kk < 4; ++kk)
      S = __builtin_amdgcn_wmma_f32_16x16x4_f32(
              false, aS[kk], false, bO[kk], (short)0, S, false, false);

    // ---- mask + LeakyReLU + scale + online max; P numerator values ----
    float pO[8];
    #pragma unroll
    for (int r = 0; r < 8; ++r) {
      float s = S[r] * av[r];
      s = (s >= 0.f) ? s : 0.2f * s;       // LeakyReLU(0.2)
      s = s * 0.25f;                       // 1/sqrt(ATT=16)
      s = (av[r] != 0.f) ? s : -1.0e9f;    // additive mask

      // row-wise max across the 16 lanes of this half
      float mx = s;
      mx = fmaxf(mx, __shfl_xor(mx, 1));
      mx = fmaxf(mx, __shfl_xor(mx, 2));
      mx = fmaxf(mx, __shfl_xor(mx, 4));
      mx = fmaxf(mx, __shfl_xor(mx, 8));
      const float mnew  = fmaxf(m[r], mx);
      const float scale = __expf(m[r] - mnew);
      const float pe    = __expf(s - mnew);
      m[r]  = mnew;
      O[r]  = O[r]  * scale;
      Lm[r] = Lm[r] * scale;
      pO[r] = (av[r] != 0.f) ? pe : 0.f;   // numerator re-masked (ref attn*mask)
    }

    // ---- re-layout P: C-layout -> LDS [M][K] -> A-layout ----
    #pragma unroll
    for (int r = 0; r < 8; ++r)
      pbuf[wv][r + 8 * half][lm] = pO[r];
    // same-wave LDS ops are in-order; compiler inserts s_wait_dscnt for reads
    v2f aP[4];
    #pragma unroll
    for (int kk = 0; kk < 4; ++kk) {
      const int k0 = kk * 4 + 2 * half;
      aP[kk] = *(const v2f*)(&pbuf[wv][lm][k0]);
    }

    // ---- B operand: feats column tile (K = col node, N = out channel) ----
    v2f bF[4];
    #pragma unroll
    for (int kk = 0; kk < 4; ++kk) {
      const int k0 = kk * 4 + 2 * half;
      const int n0 = min(colbase + k0,     N_ - 1);
      const int n1 = min(colbase + k0 + 1, N_ - 1);
      v2f t;
      t.x = ft[(size_t)n0 * 16 + lm];
      t.y = ft[(size_t)n1 * 16 + lm];
      bF[kk] = t;
    }

    // ---- O += P @ feats_tile ;  L += P @ ones (row sums) ----
    #pragma unroll
    for (int kk = 0; kk < 4; ++kk)
      O = __builtin_amdgcn_wmma_f32_16x16x4_f32(
              false, aP[kk], false, bF[kk], (short)0, O, false, false);
    #pragma unroll
    for (int kk = 0; kk < 4; ++kk)
      Lm = __builtin_amdgcn_wmma_f32_16x16x4_f32(
              false, aP[kk], false, onesB, (short)0, Lm, false, false);
  }

  // ---- normalize, bias, relu, store [B,N,H*16] (divergence OK: no more wmma) ----
  const float bv = bias[hd * 16 + lm];
  #pragma unroll
  for (int r = 0; r < 8; ++r) {
    const int rn = rt * 16 + r + 8 * half;
    if (rn < N_) {
      // L==0 <=> node has no neighbors: reference yields relu(bias) there.
      float v = (Lm[r] > 0.f) ? (O[r] / Lm[r]) : 0.f;
      v = fmaxf(v + bv, 0.f);
      out[((size_t)b * N_ + rn) * (H_ * OUT_) + hd * OUT_ + lm] = v;
    }
  }
}

// ---------------------------------------------------------------------------
extern "C" void kernel_launch(void* const* d_in, const int* in_sizes, int n_in,
                              void* d_out, int out_size, void* d_ws, size_t ws_size,
                              hipStream_t stream) {
  const float* h    = (const float*)d_in[0];  // [B,N,F]
  const float* a    = (const float*)d_in[1];  // [B,N,N]
  const float* Wk   = (const float*)d_in[2];  // [H,F,OUT]
  const float* Wa1  = (const float*)d_in[3];  // [H,F,ATT]
  const float* Wa2  = (const float*)d_in[4];  // [H,F,ATT]
  const float* bias = (const float*)d_in[5];  // [1,1,H,OUT]
  float* out = (float*)d_out;

  float* ws     = (float*)d_ws;
  const size_t projN = (size_t)B_ * H_ * N_ * 16;   // 384000 floats each
  float* feats  = ws;
  float* fself  = ws + projN;
  float* fother = ws + 2 * projN;

  gat_project<<<(B_ * N_ * H_ + 255) / 256, 256, 0, stream>>>(
      h, Wk, Wa1, Wa2, feats, fself, fother);
  gat_attention<<<NT_, 256, 0, stream>>>(a, feats, fself, fother, bias, out);
}